// GNNPolicy_19189913878709
// MI455X (gfx1250) — compile-verified
//
#include <hip/hip_runtime.h>
#include <math.h>

typedef __attribute__((ext_vector_type(2))) float v2f;
typedef __attribute__((ext_vector_type(8))) float v8f;

#define HD 64
#define BN_EPS 1e-5f
#define ROWS_PER_STAT_BLK 1024

// LDS strides (padded for bank-conflict avoidance under the half-wave K split)
#define LDA64 41   // A-transposed [k][m], m = 0..31
#define LDB64 72   // B [k][n], n = 0..63  (72*4 = 288 bytes/row, 16B aligned)
#define LDT0 17    // mlp t0 transposed [k=64][m=16]
#define LDT1 33    // mlp t1 row-major [m=16][j=32]
#define LDW1 40    // Wm1 [k=64][n=32]   (40*4 = 160 bytes/row, 16B aligned)

static __device__ __forceinline__ void atomic_add_f32(float* p, float v) {
  unsafeAtomicAdd(p, v);  // global_atomic_add_f32 (non-returning HW float atomic)
}

// LDS byte offset of a __shared__ element: generic LDS pointers on gfx1250 are
// {shared_aperture_hi32, lds_byte_offset32} (ISA 10.2), so the low 32 bits are
// exactly the wave-relative LDS address the async-DMA VDST operand expects.
static __device__ __forceinline__ unsigned lds_addr_of(const void* p) {
  return (unsigned)(unsigned long long)(size_t)p;
}

// Async DMA one 16-byte chunk global -> LDS (ASYNCcnt-tracked, bypasses VGPRs).
static __device__ __forceinline__ void async_copy_b128(unsigned lds_byte, const float* gsrc) {
  asm volatile("global_load_async_to_lds_b128 %0, %1, off"
               :: "v"(lds_byte), "v"(gsrc)
               : "memory");
}
static __device__ __forceinline__ void wait_async_zero() {
  asm volatile("s_wait_asynccnt 0x0" ::: "memory");
}

// One wave computes a 16x16 f32 tile: acc += A(16xK) * B(Kx16) using
// V_WMMA_F32_16X16X4_F32. aT is column-of-A-major: aT[k*lda + m] = A[m][k].
// b is row-major: b[k*ldb + n]. Caller pre-offsets aT by the row tile and b by
// the col tile. Per-lane fragments: lanes 0-15 carry (m=lane, k=0,1),
// lanes 16-31 carry (m=lane-16, k=2,3) within each K=4 step (ISA 7.12.2).
template <int K, int LDAT, int LDB>
static __device__ __forceinline__ v8f wmma_tile_f32(const float* aT, const float* b, v8f acc) {
  const int lane = threadIdx.x & 31;
  const int mn   = lane & 15;           // A row for src0, B col for src1
  const int kb   = (lane >> 4) << 1;    // 0 or 2
#pragma unroll
  for (int k = 0; k < K; k += 4) {
    v2f av, bv;
    av.x = aT[(k + kb    ) * LDAT + mn];
    av.y = aT[(k + kb + 1) * LDAT + mn];
    bv.x = b [(k + kb    ) * LDB  + mn];
    bv.y = b [(k + kb + 1) * LDB  + mn];
    acc = __builtin_amdgcn_wmma_f32_16x16x4_f32(false, av, false, bv, (short)0, acc,
                                                false, false);
  }
  return acc;
}

// ---------------- degree / norm ----------------
__global__ void k_init_deg(float* degf, int n) {
  int i = blockIdx.x * blockDim.x + threadIdx.x;
  if (i < n) degf[i] = 1.0f;  // self-loop
}

__global__ void k_count_deg(const int* __restrict__ dst, float* degf, int e) {
  int i = blockIdx.x * blockDim.x + threadIdx.x;
  if (i < e) atomic_add_f32(&degf[dst[i]], 1.0f);
}

__global__ void k_dinv(const float* __restrict__ degf, float* __restrict__ dinv, int n) {
  int i = blockIdx.x * blockDim.x + threadIdx.x;
  if (i < n) dinv[i] = rsqrtf(degf[i]);
}

// ---------------- GEMM: h = x[N,4] @ W0[4,64] (K=4, single WMMA/tile) ----------------
__global__ void k_gemm_xw0(const float* __restrict__ x, const float* __restrict__ W0,
                           float* __restrict__ out, int n) {
  const int wave = threadIdx.x >> 5;   // 0..7
  const int lane = threadIdx.x & 31;
  const int rowBlk  = blockIdx.x * 32;
  const int rowTile = rowBlk + ((wave >> 2) << 4);
  const int colTile = (wave & 3) << 4;
  if (rowBlk + 32 <= n) {
    const int m  = lane & 15;
    const int kb = (lane >> 4) << 1;   // 0 or 2
    const int row = rowTile + m;
    v2f av = ((const v2f*)(x + (size_t)row * 4))[kb >> 1];  // x[row][kb], x[row][kb+1]
    v2f bv;
    bv.x = W0[(kb    ) * HD + colTile + m];
    bv.y = W0[(kb + 1) * HD + colTile + m];
    v8f acc = {};
    acc = __builtin_amdgcn_wmma_f32_16x16x4_f32(false, av, false, bv, (short)0, acc,
                                                false, false);
    const int mhi = (lane >> 4) << 3;
#pragma unroll
    for (int r = 0; r < 8; ++r)
      out[(size_t)(rowTile + mhi + r) * HD + colTile + m] = acc[r];
  } else {
    for (int idx = threadIdx.x; idx < 32 * HD; idx += blockDim.x) {
      int r = rowBlk + (idx >> 6), c = idx & 63;
      if (r < n) {
        float s = 0.f;
        for (int k = 0; k < 4; ++k) s += x[(size_t)r * 4 + k] * W0[k * HD + c];
        out[(size_t)r * HD + c] = s;
      }
    }
  }
}

// ---------------- aggregation: self-loop init (+bias), then edge scatter ----------------
__global__ void k_agg_init(const float* __restrict__ hlin, const float* __restrict__ dinv,
                           const float* __restrict__ bias, float* __restrict__ out, int n) {
  size_t i = (size_t)blockIdx.x * blockDim.x + threadIdx.x;
  if (i < (size_t)n * HD) {
    int node = (int)(i >> 6), f = (int)(i & 63);
    float di = dinv[node];
    out[i] = hlin[i] * di * di + bias[f];
  }
}

// one wave per edge; 32 lanes x float2 = 64 features
__global__ void k_agg_edges(const float* __restrict__ hlin, const int* __restrict__ src,
                            const int* __restrict__ dst, const float* __restrict__ dinv,
                            float* __restrict__ out, int e) {
  long long gid = (long long)blockIdx.x * blockDim.x + threadIdx.x;
  int edge = (int)(gid >> 5);
  int lane = (int)(gid & 31);
  if (edge < e) {
    int s = src[edge], d = dst[edge];
    float nrm = dinv[s] * dinv[d];
    float2 v = ((const float2*)(hlin + (size_t)s * HD))[lane];
    float* op = out + (size_t)d * HD + lane * 2;
    atomic_add_f32(op,     v.x * nrm);
    atomic_add_f32(op + 1, v.y * nrm);
  }
}

// ---------------- batchnorm ----------------
__global__ void k_zero(float* p, int n) {
  int i = blockIdx.x * blockDim.x + threadIdx.x;
  if (i < n) p[i] = 0.f;
}

__global__ void k_bn_stats(const float* __restrict__ x, float* __restrict__ stats, int n) {
  __shared__ float ls[2][4][HD];
  const int f = threadIdx.x & 63;
  const int rsub = threadIdx.x >> 6;  // 0..3
  float s = 0.f, ss = 0.f;
  int row0 = blockIdx.x * ROWS_PER_STAT_BLK;
  int rowEnd = min(n, row0 + ROWS_PER_STAT_BLK);
  for (int r = row0 + rsub; r < rowEnd; r += 4) {
    float v = x[(size_t)r * HD + f];
    s += v; ss += v * v;
  }
  ls[0][rsub][f] = s;
  ls[1][rsub][f] = ss;
  __syncthreads();
  if (threadIdx.x < HD) {
    float a = ls[0][0][f] + ls[0][1][f] + ls[0][2][f] + ls[0][3][f];
    float b = ls[1][0][f] + ls[1][1][f] + ls[1][2][f] + ls[1][3][f];
    atomic_add_f32(&stats[f], a);
    atomic_add_f32(&stats[HD + f], b);
  }
}

__global__ void k_bn_apply_relu(float* __restrict__ x, const float* __restrict__ stats,
                                const float* __restrict__ g, const float* __restrict__ be,
                                int n) {
  size_t i = (size_t)blockIdx.x * blockDim.x + threadIdx.x;
  if (i < (size_t)n * HD) {
    int f = (int)(i & 63);
    float invN = 1.0f / (float)n;
    float mu  = stats[f] * invN;
    float var = stats[HD + f] * invN - mu * mu;   // biased variance
    float sc = g[f] * rsqrtf(var + BN_EPS);
    float sh = be[f] - mu * sc;
    float v = x[i] * sc + sh;
    x[i] = v > 0.f ? v : 0.f;
  }
}

// ---------------- GEMM: out[N,64] = in[N,64] @ W[64,64] (WMMA, LDS-staged) ----------------
__global__ void k_gemm64(const float* __restrict__ in, const float* __restrict__ W,
                         float* __restrict__ out, int n) {
  __shared__ __align__(16) float At[HD * LDA64];  // [k][m], m in 0..31 (A transposed)
  __shared__ __align__(16) float Bs[HD * LDB64];  // [k][n]
  const int tid = threadIdx.x;      // 256 threads = 8 waves
  // Async DMA W -> LDS (1024 x b128 chunks; 4 per thread), overlapped with the
  // VALU/DS transpose staging of A below.
  for (int i = tid; i < (HD * HD) / 4; i += 256) {
    int k = i >> 4;            // row (16 chunks per 64-float row)
    int c = (i & 15) << 2;     // col, 4-float aligned -> 16B aligned both sides
    async_copy_b128(lds_addr_of(&Bs[k * LDB64 + c]), W + k * HD + c);
  }
  const int rowBlk = blockIdx.x * 32;
  if (rowBlk + 32 <= n) {
    for (int i = tid; i < 32 * HD; i += 256) {
      int m = i >> 6, k = i & 63;   // consecutive tid -> consecutive k (coalesced read)
      At[k * LDA64 + m] = in[(size_t)(rowBlk + m) * HD + k];
    }
    wait_async_zero();   // this wave's W chunks are in LDS
    __syncthreads();     // => all waves' chunks are in LDS
    const int wave = tid >> 5, lane = tid & 31;
    const int mt = (wave >> 2) << 4;  // 0 or 16
    const int nt = (wave & 3) << 4;
    v8f acc = {};
    acc = wmma_tile_f32<HD, LDA64, LDB64>(At + mt, Bs + nt, acc);
    const int m = lane & 15;
    const int mhi = (lane >> 4) << 3;
#pragma unroll
    for (int r = 0; r < 8; ++r)
      out[(size_t)(rowBlk + mt + mhi + r) * HD + nt + m] = acc[r];
  } else {
    wait_async_zero();
    __syncthreads();
    for (int idx = tid; idx < 32 * HD; idx += 256) {
      int r = rowBlk + (idx >> 6), c = idx & 63;
      if (r < n) {
        float s = 0.f;
        for (int k = 0; k < HD; ++k) s += in[(size_t)r * HD + k] * Bs[k * LDB64 + c];
        out[(size_t)r * HD + c] = s;
      }
    }
  }
}

// ---------------- fused MLP head: relu(h@Wm0+bm0) -> relu(@Wm1+bm1) -> sigmoid(@Wm2+bm2) ----
__global__ void k_mlp(const float* __restrict__ in,
                      const float* __restrict__ Wm0, const float* __restrict__ bm0,
                      const float* __restrict__ Wm1, const float* __restrict__ bm1,
                      const float* __restrict__ Wm2, const float* __restrict__ bm2,
                      float* __restrict__ out, int n) {
  __shared__ __align__(16) float W0s[HD * LDB64];   // [k][n], 64x64
  __shared__ __align__(16) float W1s[HD * LDW1];    // [k][n], 64x32
  __shared__ float At [HD * LDT0];    // input rows transposed [k=64][m=16]
  __shared__ float t0t[HD * LDT0];    // t0 transposed [k=64][m=16]
  __shared__ float t1s[16 * LDT1];    // t1 row-major [m=16][j=32]
  const int tid = threadIdx.x;        // 128 threads = 4 waves
  const int wave = tid >> 5, lane = tid & 31;
  // Async DMA both weight panels -> LDS.
  for (int i = tid; i < (HD * HD) / 4; i += 128) {       // Wm0: 1024 chunks
    int k = i >> 4, c = (i & 15) << 2;
    async_copy_b128(lds_addr_of(&W0s[k * LDB64 + c]), Wm0 + k * HD + c);
  }
  for (int i = tid; i < (HD * 32) / 4; i += 128) {       // Wm1: 512 chunks
    int k = i >> 3, c = (i & 7) << 2;
    async_copy_b128(lds_addr_of(&W1s[k * LDW1 + c]), Wm1 + k * 32 + c);
  }
  const int rowBlk = blockIdx.x * 16;
  if (rowBlk + 16 <= n) {
    for (int i = tid; i < 16 * HD; i += 128) {
      int m = i >> 6, k = i & 63;
      At[k * LDT0 + m] = in[(size_t)(rowBlk + m) * HD + k];
    }
    wait_async_zero();
    __syncthreads();
    {  // stage 1: t0 = relu(in @ Wm0 + bm0); 4 waves -> 4 col tiles; store transposed
      const int n0 = wave << 4;
      v8f acc = {};
      acc = wmma_tile_f32<HD, LDT0, LDB64>(At, W0s + n0, acc);
      const int nn = lane & 15, mhi = (lane >> 4) << 3;
      const float bias = bm0[n0 + nn];
#pragma unroll
      for (int r = 0; r < 8; ++r) {
        float v = acc[r] + bias;
        t0t[(n0 + nn) * LDT0 + mhi + r] = v > 0.f ? v : 0.f;
      }
    }
    __syncthreads();
    if (wave < 2) {  // stage 2: t1 = relu(t0 @ Wm1 + bm1); 2 waves -> 2 col tiles
      const int n0 = wave << 4;
      v8f acc = {};
      acc = wmma_tile_f32<HD, LDT0, LDW1>(t0t, W1s + n0, acc);
      const int nn = lane & 15, mhi = (lane >> 4) << 3;
      const float bias = bm1[n0 + nn];
#pragma unroll
      for (int r = 0; r < 8; ++r) {
        float v = acc[r] + bias;
        t1s[(mhi + r) * LDT1 + n0 + nn] = v > 0.f ? v : 0.f;
      }
    }
    __syncthreads();
    if (tid < 16) {  // stage 3: sigmoid(t1 @ Wm2 + bm2)
      float s = bm2[0];
#pragma unroll
      for (int j = 0; j < 32; ++j) s += t1s[tid * LDT1 + j] * Wm2[j];
      out[rowBlk + tid] = 1.0f / (1.0f + expf(-s));
    }
  } else {
    wait_async_zero();
    __syncthreads();
    if (tid < 16 && rowBlk + tid < n) {
      const int row = rowBlk + tid;
      float t0[HD], t1[32];
      for (int c = 0; c < HD; ++c) {
        float s = bm0[c];
        for (int k = 0; k < HD; ++k) s += in[(size_t)row * HD + k] * W0s[k * LDB64 + c];
        t0[c] = s > 0.f ? s : 0.f;
      }
      for (int c = 0; c < 32; ++c) {
        float s = bm1[c];
        for (int k = 0; k < HD; ++k) s += t0[k] * W1s[k * LDW1 + c];
        t1[c] = s > 0.f ? s : 0.f;
      }
      float s = bm2[0];
      for (int j = 0; j < 32; ++j) s += t1[j] * Wm2[j];
      out[row] = 1.0f / (1.0f + expf(-s));
    }
  }
}

// ---------------- host launch ----------------
extern "C" void kernel_launch(void* const* d_in, const int* in_sizes, int n_in,
                              void* d_out, int out_size, void* d_ws, size_t ws_size,
                              hipStream_t stream) {
  const float* x   = (const float*)d_in[0];
  const int*   src = (const int*)  d_in[1];
  const int*   dst = (const int*)  d_in[2];
  const float* W0  = (const float*)d_in[3];
  const float* b0  = (const float*)d_in[4];
  const float* W1  = (const float*)d_in[5];
  const float* b1  = (const float*)d_in[6];
  const float* g0  = (const float*)d_in[7];
  const float* be0 = (const float*)d_in[8];
  const float* g1  = (const float*)d_in[9];
  const float* be1 = (const float*)d_in[10];
  const float* Wm0 = (const float*)d_in[11];
  const float* bm0 = (const float*)d_in[12];
  const float* Wm1 = (const float*)d_in[13];
  const float* bm1 = (const float*)d_in[14];
  const float* Wm2 = (const float*)d_in[15];
  const float* bm2 = (const float*)d_in[16];

  const int n = in_sizes[0] / 4;
  const int e = in_sizes[1];

  // workspace layout
  float* ws    = (float*)d_ws;
  float* degf  = ws;                       // n
  float* dinv  = degf + n;                 // n
  float* stats = dinv + n;                 // 128
  float* bufA  = stats + 128;              // n*64
  float* bufB  = bufA + (size_t)n * HD;    // n*64

  const int T = 256;
  const int gN    = (n + T - 1) / T;
  const int gE    = (e + T - 1) / T;
  const int gNH   = (int)(((size_t)n * HD + T - 1) / T);
  const int gRow32 = (n + 31) / 32;
  const int gRow16 = (n + 15) / 16;
  const int gEdgeW = (int)(((long long)e * 32 + T - 1) / T);
  const int gStat  = (n + ROWS_PER_STAT_BLK - 1) / ROWS_PER_STAT_BLK;

  float* outp = (float*)d_out;

  // degree + symmetric normalization
  k_init_deg <<<gN, T, 0, stream>>>(degf, n);
  k_count_deg<<<gE, T, 0, stream>>>(dst, degf, e);
  k_dinv     <<<gN, T, 0, stream>>>(degf, dinv, n);

  // --- GCN layer 1 ---
  k_gemm_xw0 <<<gRow32, T, 0, stream>>>(x, W0, bufA, n);
  k_agg_init <<<gNH, T, 0, stream>>>(bufA, dinv, b0, bufB, n);
  k_agg_edges<<<gEdgeW, T, 0, stream>>>(bufA, src, dst, dinv, bufB, e);
  k_zero     <<<1, 128, 0, stream>>>(stats, 128);
  k_bn_stats <<<gStat, T, 0, stream>>>(bufB, stats, n);
  k_bn_apply_relu<<<gNH, T, 0, stream>>>(bufB, stats, g0, be0, n);

  // --- GCN layer 2 ---
  k_gemm64   <<<gRow32, T, 0, stream>>>(bufB, W1, bufA, n);
  k_agg_init <<<gNH, T, 0, stream>>>(bufA, dinv, b1, bufB, n);
  k_agg_edges<<<gEdgeW, T, 0, stream>>>(bufA, src, dst, dinv, bufB, e);
  k_zero     <<<1, 128, 0, stream>>>(stats, 128);
  k_bn_stats <<<gStat, T, 0, stream>>>(bufB, stats, n);
  k_bn_apply_relu<<<gNH, T, 0, stream>>>(bufB, stats, g1, be1, n);

  // --- MLP head (fused, WMMA) ---
  k_mlp<<<gRow16, 128, 0, stream>>>(bufB, Wm0, bm0, Wm1, bm1, Wm2, bm2, outp, n);
}